// MetaUpscale_59072980189500
// MI455X (gfx1250) — compile-verified
//
#include <hip/hip_runtime.h>
#include <stdint.h>

// MetaUpscale (N=2, C=64, H=W=128, S=2) for gfx1250 / MI455X.
//
// Roofline: lw stream = 453 MB read exactly once -> ~19.4 us at 23.3 TB/s;
// only ~23 TFLOPS fp32 needed -> pure HBM streaming kernel.
//  * lw: non-temporal b64 loads (read-once; don't pollute 192 MB L2)
//  * x 3x3 patch: staged to LDS by the Tensor Data Mover (tensor_load_to_lds,
//    TENSORcnt) for interior blocks; per-lane async global->LDS b32 + zero
//    fill (ASYNCcnt) for the ~4.6% border blocks.
//  * wave32 shuffle reduction of the 6 (n,c) accumulators.
//
// out[n,c,oy,ox] = sum_k patch(x)[n, oy/2, ox/2, k] * lw[oy, ox, k, c]
//   k = ci*9 + i*3 + j,  patch value = x[n, ci, oy/2-1+i, ox/2-1+j] (zero pad)

typedef float v2f  __attribute__((ext_vector_type(2)));
typedef uint32_t u32x4 __attribute__((ext_vector_type(4)));
typedef uint32_t u32x8 __attribute__((ext_vector_type(8)));

__device__ __forceinline__ v2f nt_load2(const float* p) {
  return __builtin_nontemporal_load((const v2f*)p);
}

__global__ __launch_bounds__(256) void meta_upscale_kernel(
    const float* __restrict__ x,    // [2, 64, 128, 128]
    const float* __restrict__ lw,   // [256, 256, 576, 3]
    float* __restrict__ out)        // [2, 3, 256, 256]
{
  // Patch layout (TDM tile order, X fastest): [n(2)][ci(64)][i(3)][jj(4)]
  // jj indexes global column (wb-1+jj); the two input pixels share it.
  __shared__ __align__(16) float patch[1536];

  const int tid = threadIdx.x;
  const int h0  = blockIdx.x >> 6;         // input row 0..127
  const int wb  = (blockIdx.x & 63) << 1;  // first of two input cols

  const bool interior = (h0 >= 1) & (h0 <= 126) & (wb >= 2) & (wb <= 124);

  if (interior) {
    // ---- Tensor Data Mover: one 3D-tile DMA per batch image ----
    if (tid < 32) {                        // one wave issues; EXEC ignored by TDM
      const uint32_t lds_base = (uint32_t)(uintptr_t)patch;
#pragma unroll
      for (int n = 0; n < 2; ++n) {
        const uint64_t ga =
            (uint64_t)(uintptr_t)(x + (((size_t)n * 64) * 128 + (h0 - 1)) * 128 + (wb - 1));
        // D# group 0: count=1 | lds_addr | global_addr | type=2 (bits 127:126)
        u32x4 g0 = { 0x1u,
                     lds_base + (uint32_t)n * 3072u,
                     (uint32_t)ga,
                     (uint32_t)(ga >> 32) | 0x80000000u };
        // D# group 1: data_size=2 (4B); tensor_dim0=128 @bits79:48;
        // tensor_dim1=128 @111:80; tile_dim0=4 @127:112; tile_dim1=3,
        // tile_dim2=64 @159:128; stride0=128 @207:160; stride1=16384 @255:208
        u32x8 g1 = { 0x00020000u,          // workgroup_mask=0, data_size=4B
                     128u << 16,           // tensor_dim0[15:0] in [31:16]
                     128u << 16,           // dim0[31:16]=0 | tensor_dim1[15:0]
                     4u << 16,             // dim1[31:16]=0 | tile_dim0=4
                     3u | (64u << 16),     // tile_dim1=3 | tile_dim2=64
                     128u,                 // tensor_dim0_stride[31:0]
                     16384u << 16,         // s0[47:32]=0 | s1[15:0]=16384
                     0u };                 // tensor_dim1_stride[47:16]
        // D# group 2: tensor_dim2=64 | (dim3 unused) | stride2=16384 | tile_dim3=0
        u32x4 g2 = { 64u, 0u, 16384u, 0u };
        asm volatile("tensor_load_to_lds %0, %1, %2, %3"
                     :: "s"(g0), "s"(g1), "s"(g2), "s"(g2) : "memory");
      }
      __builtin_amdgcn_s_wait_tensorcnt(0);
    }
  } else {
    // ---- Border blocks: per-lane async global->LDS + zero padding ----
#pragma unroll
    for (int n = 0; n < 2; ++n) {
#pragma unroll
      for (int t = 0; t < 3; ++t) {
        int e    = tid + t * 256;          // 0..767 == (ci*3+i)*4 + jj
        int jj   = e & 3;
        int rem  = e >> 2;                 // ci*3 + i
        int ci   = rem / 3;
        int i    = rem - ci * 3;
        int gy   = h0 - 1 + i;
        int gx   = wb - 1 + jj;
        int lidx = n * 768 + e;
        if ((unsigned)gy < 128u && (unsigned)gx < 128u) {
          const float* src = x + ((((size_t)n * 64 + ci) * 128 + gy) << 7) + gx;
          uint32_t ldsoff = (uint32_t)(uintptr_t)(patch + lidx);
          asm volatile("global_load_async_to_lds_b32 %0, %1, off"
                       :: "v"(ldsoff), "v"(src) : "memory");
        } else {
          patch[lidx] = 0.0f;              // zero padding (disjoint addresses)
        }
      }
    }
    asm volatile("s_wait_asynccnt 0x0" ::: "memory");
  }
  __syncthreads();

  // ---- Each wave owns one output pixel; stream its 1728-float lw block ----
  const int lane = tid & 31;
  const int wv   = tid >> 5;               // 0..7
  const int wp   = wv >> 2;                // which of the two input pixels
  const int si   = (wv >> 1) & 1;
  const int sj   = wv & 1;
  const int oy   = (h0 << 1) + si;
  const int ox   = ((wb + wp) << 1) + sj;

  // Lane l handles rows k = 2*(32*m + l), k+1 ; 6 contiguous dwords per step.
  const float* lwp = lw + ((size_t)((oy << 8) + ox)) * 1728 + lane * 6;

  int k0 = lane << 1;
  int ci = k0 / 9;
  int r  = k0 - ci * 9;                    // k % 9 ; updated incrementally

  float a00 = 0.f, a01 = 0.f, a02 = 0.f;   // n=0, c=0..2
  float a10 = 0.f, a11 = 0.f, a12 = 0.f;   // n=1, c=0..2

#pragma unroll
  for (int m = 0; m < 9; ++m) {
    // Two rows of weights: 3 aligned non-temporal b64 loads (read-once stream).
    v2f q01 = nt_load2(lwp);        // w(k,0)   w(k,1)
    v2f q23 = nt_load2(lwp + 2);    // w(k,2)   w(k+1,0)
    v2f q45 = nt_load2(lwp + 4);    // w(k+1,1) w(k+1,2)

    // patch index: ci*12 + (i*4 + j) + wp ; i*4+j = r + r/3, r/3 = (r*11)>>5
    int idxA = ci * 12 + r + ((r * 11) >> 5) + wp;
    int r1 = r + 1, c1 = ci;
    if (r1 == 9) { r1 = 0; ++c1; }
    int idxB = c1 * 12 + r1 + ((r1 * 11) >> 5) + wp;

    float xa0 = patch[idxA];        // n=0, row k
    float xa1 = patch[idxA + 768];  // n=1, row k
    float xb0 = patch[idxB];        // n=0, row k+1
    float xb1 = patch[idxB + 768];  // n=1, row k+1

    a00 = fmaf(xa0, q01.x, a00); a01 = fmaf(xa0, q01.y, a01); a02 = fmaf(xa0, q23.x, a02);
    a10 = fmaf(xa1, q01.x, a10); a11 = fmaf(xa1, q01.y, a11); a12 = fmaf(xa1, q23.x, a12);
    a00 = fmaf(xb0, q23.y, a00); a01 = fmaf(xb0, q45.x, a01); a02 = fmaf(xb0, q45.y, a02);
    a10 = fmaf(xb1, q23.y, a10); a11 = fmaf(xb1, q45.x, a11); a12 = fmaf(xb1, q45.y, a12);

    lwp += 192;                       // advance 64 rows * 3 floats
    r += 1; ci += 7;                  // k += 64 : 64 = 7*9 + 1
    if (r >= 9) { r -= 9; ++ci; }
  }

  // ---- wave32 cross-lane reduction ----
#pragma unroll
  for (int off = 16; off > 0; off >>= 1) {
    a00 += __shfl_xor(a00, off, 32);
    a01 += __shfl_xor(a01, off, 32);
    a02 += __shfl_xor(a02, off, 32);
    a10 += __shfl_xor(a10, off, 32);
    a11 += __shfl_xor(a11, off, 32);
    a12 += __shfl_xor(a12, off, 32);
  }

  if (lane == 0) {
    int p = (oy << 8) + ox;           // oy*256 + ox
    out[p]          = a00;            // n=0,c=0
    out[p + 65536]  = a01;            // n=0,c=1
    out[p + 131072] = a02;            // n=0,c=2
    out[p + 196608] = a10;            // n=1,c=0
    out[p + 262144] = a11;            // n=1,c=1
    out[p + 327680] = a12;            // n=1,c=2
  }
}

extern "C" void kernel_launch(void* const* d_in, const int* in_sizes, int n_in,
                              void* d_out, int out_size, void* d_ws, size_t ws_size,
                              hipStream_t stream) {
  (void)in_sizes; (void)n_in; (void)d_ws; (void)ws_size; (void)out_size;
  const float* x  = (const float*)d_in[0];
  const float* lw = (const float*)d_in[1];
  float* out      = (float*)d_out;
  // 128 input rows * 64 column-pairs ; 8 waves/block -> one output pixel each.
  dim3 grid(128 * 64), block(256);
  hipLaunchKernelGGL(meta_upscale_kernel, grid, block, 0, stream, x, lw, out);
}